// fab_penalty_ls_gap_72602127171971
// MI455X (gfx1250) — compile-verified
//
#include <hip/hip_runtime.h>
#include <math.h>

#define NRHO1 32
#define NRHO  (NRHO1 * NRHO1)   // 1024 control points
#define NXPHI 480               // rows of phi
#define NYPHI 240               // cols of phi before mirroring
#define NFULL 480               // cols after mirroring

typedef __attribute__((ext_vector_type(2))) float v2f;
typedef __attribute__((ext_vector_type(8))) float v8f;

// ---------------------------------------------------------------------------
// Stage A1: K[c,c'] = exp(-(dy^2+dx^2)/(2 sig^2)), c = i*32+j over (y_rho[i], x_rho[j])
// ---------------------------------------------------------------------------
__global__ void build_k_kernel(float* __restrict__ K,
                               const float* __restrict__ x_rho,
                               const float* __restrict__ y_rho,
                               const float* __restrict__ rho_size) {
  int idx = blockIdx.x * blockDim.x + threadIdx.x;   // NRHO*NRHO entries
  float sig = rho_size[0];
  float inv2s2 = 1.0f / (2.0f * sig * sig);
  int c  = idx >> 10;
  int c2 = idx & (NRHO - 1);
  int i  = c >> 5,  j  = c & 31;
  int i2 = c2 >> 5, j2 = c2 & 31;
  float dy = y_rho[i] - y_rho[i2];
  float dx = x_rho[j] - x_rho[j2];
  K[idx] = expf(-(dx * dx + dy * dy) * inv2s2);
}

// ---------------------------------------------------------------------------
// Stage A2: solve K * model = params (LU without pivoting, K is SPD).
// Single workgroup of 1024 threads (one thread per row), pivot row in LDS.
// ---------------------------------------------------------------------------
__global__ void solve_kernel(float* __restrict__ K,
                             const float* __restrict__ params,
                             float* __restrict__ model) {
  const int N = NRHO;
  int tid = threadIdx.x;
  __shared__ float b_sh[NRHO];
  __shared__ float piv[NRHO];
  __shared__ float x_sh[NRHO];
  b_sh[tid] = params[tid];
  __syncthreads();

  // forward elimination
  for (int k = 0; k < N - 1; ++k) {
    piv[tid] = K[k * N + tid];           // cache pivot row
    __threadfence_block();
    __syncthreads();
    if (tid > k) {
      float m  = K[tid * N + k] / piv[k];
      float bk = b_sh[k];
      float* __restrict__ row = K + tid * N;
      for (int col = k + 1; col < N; ++col)
        row[col] -= m * piv[col];
      b_sh[tid] -= m * bk;
    }
    __threadfence_block();
    __syncthreads();
  }

  // back substitution (column-oriented)
  for (int k = N - 1; k >= 0; --k) {
    if (tid == 0) x_sh[k] = b_sh[k] / K[k * N + k];
    __syncthreads();
    if (tid < k) b_sh[tid] -= K[tid * N + k] * x_sh[k];
    __syncthreads();
  }
  model[tid] = x_sh[tid];
}

// ---------------------------------------------------------------------------
// Stage B1: separable RBF factors.
//   ExT[j*240 + b] = exp(-(x_rho[j]-x_phi[b])^2 / 2sig^2)   (32 x 240, B matrix)
//   Ey [a*32  + i] = exp(-(y_rho[i]-y_phi[a])^2 / 2sig^2)   (480 x 32, A matrix)
// ---------------------------------------------------------------------------
__global__ void build_e_kernel(float* __restrict__ ExT, float* __restrict__ Ey,
                               const float* __restrict__ x_rho,
                               const float* __restrict__ y_rho,
                               const float* __restrict__ x_phi,
                               const float* __restrict__ y_phi,
                               const float* __restrict__ rho_size) {
  int idx = blockIdx.x * blockDim.x + threadIdx.x;
  float sig = rho_size[0];
  float inv2s2 = 1.0f / (2.0f * sig * sig);
  if (idx < NYPHI * NRHO1) {
    int b = idx >> 5, j = idx & 31;
    float d = x_rho[j] - x_phi[b];
    ExT[j * NYPHI + b] = expf(-d * d * inv2s2);
  } else if (idx < NYPHI * NRHO1 + NXPHI * NRHO1) {
    int k = idx - NYPHI * NRHO1;
    int a = k >> 5, i = k & 31;
    float d = y_rho[i] - y_phi[a];
    Ey[a * NRHO1 + i] = expf(-d * d * inv2s2);
  }
}

// ---------------------------------------------------------------------------
// f32 WMMA 16x16 tile GEMM: D = A[row0.., K] x B[K, col0..]  (K multiple of 4)
// A row-major (lda), B row-major (ldb). One full wave32 per tile, EXEC all 1s.
// ISA layouts: A 16x4: lane=M, VGPR0/1=K lsb, lane-half adds K+2.
//              B 4x16: lane=N, VGPR0/1=K lsb, lane-half adds K+2.
//              D: VGPR v -> row (v + half*8), col = lane&15.
// ---------------------------------------------------------------------------
__device__ inline v8f wmma_f32_tile(const float* __restrict__ A, int lda,
                                    const float* __restrict__ B, int ldb,
                                    int K, int row0, int col0) {
  int lane = threadIdx.x & 31;
  int half = lane >> 4;
  int l16  = lane & 15;
  v8f acc = {0.f, 0.f, 0.f, 0.f, 0.f, 0.f, 0.f, 0.f};
  for (int k0 = 0; k0 < K; k0 += 4) {
    int ka = k0 + half * 2;
    v2f a, b;
    a[0] = A[(row0 + l16) * lda + ka];
    a[1] = A[(row0 + l16) * lda + ka + 1];
    b[0] = B[ka * ldb + col0 + l16];
    b[1] = B[(ka + 1) * ldb + col0 + l16];
    acc = __builtin_amdgcn_wmma_f32_16x16x4_f32(false, a, false, b,
                                                (short)0, acc, false, false);
  }
  return acc;
}

// Stage B2: T (32x240) = M (32x32, model reshaped) x ExT (32x240). 2x15 tiles.
__global__ void gemm_t_kernel(const float* __restrict__ Mmat,
                              const float* __restrict__ ExT,
                              float* __restrict__ T) {
  int tile = blockIdx.x;               // 0..29
  int row0 = (tile / 15) * 16;
  int col0 = (tile % 15) * 16;
  v8f acc = wmma_f32_tile(Mmat, NRHO1, ExT, NYPHI, NRHO1, row0, col0);
  int lane = threadIdx.x & 31, half = lane >> 4, l16 = lane & 15;
#pragma unroll
  for (int v = 0; v < 8; ++v)
    T[(row0 + v + half * 8) * NYPHI + col0 + l16] = acc[v];
}

// Stage B3: phi0 (480x240) = Ey (480x32) x T (32x240); write mirrored 480x480.
__global__ void gemm_phi_kernel(const float* __restrict__ Ey,
                                const float* __restrict__ T,
                                float* __restrict__ phi) {
  int tile = blockIdx.x;               // 0..449
  int row0 = (tile / 15) * 16;
  int col0 = (tile % 15) * 16;
  v8f acc = wmma_f32_tile(Ey, NRHO1, T, NYPHI, NRHO1, row0, col0);
  int lane = threadIdx.x & 31, half = lane >> 4, l16 = lane & 15;
#pragma unroll
  for (int v = 0; v < 8; ++v) {
    int a = row0 + v + half * 8;
    int b = col0 + l16;
    float val = acc[v];
    phi[a * NFULL + b] = val;                 // left half
    phi[a * NFULL + (NFULL - 1 - b)] = val;   // mirrored right half
  }
}

// ---------------------------------------------------------------------------
// Finite differences (central interior, one-sided borders), matching _grad1d.
// ---------------------------------------------------------------------------
__device__ inline float fd_ax0(const float* __restrict__ f, int a, int b,
                               float inv_d, float inv_2d) {
  if (a == 0)         return (f[NFULL + b] - f[b]) * inv_d;
  if (a == NXPHI - 1) return (f[a * NFULL + b] - f[(a - 1) * NFULL + b]) * inv_d;
  return (f[(a + 1) * NFULL + b] - f[(a - 1) * NFULL + b]) * inv_2d;
}
__device__ inline float fd_ax1(const float* __restrict__ f, int a, int b,
                               float inv_d, float inv_2d) {
  const float* r = f + a * NFULL;
  if (b == 0)         return (r[1] - r[0]) * inv_d;
  if (b == NFULL - 1) return (r[b] - r[b - 1]) * inv_d;
  return (r[b + 1] - r[b - 1]) * inv_2d;
}

__global__ void grad_kernel(const float* __restrict__ phi,
                            float* __restrict__ px, float* __restrict__ py,
                            const float* __restrict__ grid_size) {
  int idx = blockIdx.x * blockDim.x + threadIdx.x;   // 480*480
  int a = idx / NFULL, b = idx % NFULL;
  float d = grid_size[0];
  float inv_d = 1.0f / d, inv_2d = 0.5f / d;
  px[idx] = fd_ax0(phi, a, b, inv_d, inv_2d) + 1e-12f;
  py[idx] = fd_ax1(phi, a, b, inv_d, inv_2d) + 1e-12f;
}

__global__ void penalty_kernel(const float* __restrict__ phi,
                               const float* __restrict__ px,
                               const float* __restrict__ py,
                               float* __restrict__ partials,
                               const float* __restrict__ grid_size) {
  int idx = blockIdx.x * blockDim.x + threadIdx.x;
  int a = idx / NFULL, b = idx % NFULL;
  float d = grid_size[0];
  float inv_d = 1.0f / d, inv_2d = 0.5f / d;

  float vx  = px[idx], vy = py[idx];
  float pxx = fd_ax0(px, a, b, inv_d, inv_2d);
  float pxy = fd_ax1(px, a, b, inv_d, inv_2d);
  float pyy = fd_ax1(py, a, b, inv_d, inv_2d);
  float ph  = phi[idx];

  const float pi_d = 3.14159265358979323846f / 1.3f;  // pi/(1.3*MIN_FEATURE_SIZE)
  float pv  = fmaxf(sqrtf(vx * vx + vy * vy), 1e-8f);
  float pvv = (vx * vx * pxx + 2.0f * vx * vy * pxy + vy * vy * pyy) / (pv * pv);
  float pen = fmaxf(fabsf(pvv) / (pi_d * fabsf(ph) + 1.0f * pv) - pi_d, 0.0f);
  if (pen != pen) pen = 0.0f;   // nansum semantics

  __shared__ float red[256];
  red[threadIdx.x] = pen;
  __syncthreads();
  for (int s = 128; s > 0; s >>= 1) {
    if (threadIdx.x < s) red[threadIdx.x] += red[threadIdx.x + s];
    __syncthreads();
  }
  if (threadIdx.x == 0) partials[blockIdx.x] = red[0];
}

__global__ void reduce_kernel(const float* __restrict__ partials, int n,
                              const float* __restrict__ grid_size,
                              float* __restrict__ out) {
  __shared__ float red[1024];
  float s = 0.0f;
  for (int i = threadIdx.x; i < n; i += blockDim.x) s += partials[i];
  red[threadIdx.x] = s;
  __syncthreads();
  for (int st = 512; st > 0; st >>= 1) {
    if (threadIdx.x < st) red[threadIdx.x] += red[threadIdx.x + st];
    __syncthreads();
  }
  if (threadIdx.x == 0) {
    float d = grid_size[0];
    out[0] = red[0] * d * d;
  }
}

// ---------------------------------------------------------------------------
extern "C" void kernel_launch(void* const* d_in, const int* in_sizes, int n_in,
                              void* d_out, int out_size, void* d_ws, size_t ws_size,
                              hipStream_t stream) {
  (void)in_sizes; (void)n_in; (void)out_size; (void)ws_size;
  const float* params    = (const float*)d_in[0];
  const float* x_rho     = (const float*)d_in[1];
  const float* y_rho     = (const float*)d_in[2];
  const float* x_phi     = (const float*)d_in[3];
  const float* y_phi     = (const float*)d_in[4];
  // d_in[5], d_in[6] are nx_phi/ny_phi (compile-time constants here)
  const float* rho_size  = (const float*)d_in[7];
  const float* grid_size = (const float*)d_in[8];
  float* out = (float*)d_out;

  float* ws       = (float*)d_ws;
  float* K        = ws; ws += NRHO * NRHO;       // 4 MB
  float* model    = ws; ws += NRHO;
  float* ExT      = ws; ws += NRHO1 * NYPHI;
  float* Ey       = ws; ws += NXPHI * NRHO1;
  float* T        = ws; ws += NRHO1 * NYPHI;
  float* phi      = ws; ws += NXPHI * NFULL;
  float* px       = ws; ws += NXPHI * NFULL;
  float* py       = ws; ws += NXPHI * NFULL;
  float* partials = ws; ws += 1024;

  build_k_kernel<<<(NRHO * NRHO) / 256, 256, 0, stream>>>(K, x_rho, y_rho, rho_size);
  solve_kernel<<<1, NRHO, 0, stream>>>(K, params, model);
  build_e_kernel<<<(NYPHI * NRHO1 + NXPHI * NRHO1 + 255) / 256, 256, 0, stream>>>(
      ExT, Ey, x_rho, y_rho, x_phi, y_phi, rho_size);
  gemm_t_kernel<<<2 * 15, 32, 0, stream>>>(model, ExT, T);
  gemm_phi_kernel<<<30 * 15, 32, 0, stream>>>(Ey, T, phi);
  grad_kernel<<<(NXPHI * NFULL) / 256, 256, 0, stream>>>(phi, px, py, grid_size);
  penalty_kernel<<<(NXPHI * NFULL) / 256, 256, 0, stream>>>(phi, px, py, partials, grid_size);
  reduce_kernel<<<1, 1024, 0, stream>>>(partials, (NXPHI * NFULL) / 256, grid_size, out);
}